// PCAVProjection_20916490731597
// MI455X (gfx1250) — compile-verified
//
#include <hip/hip_runtime.h>

#define DIMD 4096
#define BROWS 2048
#define KSPLIT 8

typedef __attribute__((ext_vector_type(2))) float v2f;
typedef __attribute__((ext_vector_type(8))) float v8f;

// ---------------------------------------------------------------------------
// Kernel A: recover c from rank-1 P, compute t = z.c, zero the partial-sum buf
// ---------------------------------------------------------------------------
__global__ void pcav_prepare(const float* __restrict__ P,
                             const float* __restrict__ z,
                             float* __restrict__ c,
                             float* __restrict__ s,
                             float* __restrict__ tbuf) {
    __shared__ float sval[256];
    __shared__ int   sidx[256];
    __shared__ float s_scale;
    __shared__ int   s_j0;
    const int tid = threadIdx.x;

    // 1) argmax of diag(P) -> pivot j0 (c_j0^2 is max, guaranteed > 0)
    float best = -1.0f; int bi = 0;
    for (int j = tid; j < DIMD; j += 256) {
        float dv = P[(size_t)j * DIMD + j];
        if (dv > best) { best = dv; bi = j; }
    }
    sval[tid] = best; sidx[tid] = bi;
    __syncthreads();
    for (int off = 128; off > 0; off >>= 1) {
        if (tid < off && sval[tid + off] > sval[tid]) {
            sval[tid] = sval[tid + off]; sidx[tid] = sidx[tid + off];
        }
        __syncthreads();
    }
    if (tid == 0) { s_j0 = sidx[0]; s_scale = rsqrtf(sval[0]); }
    __syncthreads();
    const int   j0    = s_j0;
    const float scale = s_scale;

    // 2) c = P[:, j0] * rsqrt(P[j0,j0]);  3) t = sum_j z[j]*c[j]
    float tpart = 0.0f;
    for (int j = tid; j < DIMD; j += 256) {
        float cj = P[(size_t)j * DIMD + j0] * scale;
        c[j] = cj;
        tpart += z[j] * cj;
    }
    sval[tid] = tpart;
    __syncthreads();
    for (int off = 128; off > 0; off >>= 1) {
        if (tid < off) sval[tid] += sval[tid + off];
        __syncthreads();
    }
    if (tid == 0) tbuf[0] = sval[0];

    // 4) zero partial-sum buffer (ws is poisoned by the harness)
    for (int j = tid; j < BROWS; j += 256) s[j] = 0.0f;
}

// ---------------------------------------------------------------------------
// Kernel B: s[b] += x[b,:].c  via V_WMMA_F32_16X16X4_F32, K split 8 ways.
// A = 16x4 tile of x (documented layout: lane holds K = {2*(lane/16), +1}),
// B = c broadcast into all 16 columns (same K indexing), so every column of
// the 16x16 f32 accumulator carries the 16 row-dot partial sums.
// ---------------------------------------------------------------------------
__global__ void __launch_bounds__(32)
pcav_gemv_wmma(const float* __restrict__ x,
               const float* __restrict__ c,
               float* __restrict__ s) {
    const int lane  = threadIdx.x;            // one wave32 per block; EXEC all 1s
    const int rbase = blockIdx.x * 16;        // 16-row tile
    const int kbase = blockIdx.y * (DIMD / KSPLIT);
    const int kend  = kbase + (DIMD / KSPLIT);
    const int row   = rbase + (lane & 15);
    const int koff  = (lane >> 4) * 2;        // A/B f32 K placement per lane half

    const float* xrow = x + (size_t)row * DIMD + koff;
    const float* cp   = c + koff;

    v8f acc = {0.f, 0.f, 0.f, 0.f, 0.f, 0.f, 0.f, 0.f};
    for (int k0 = kbase; k0 < kend; k0 += 4) {
        v2f a = *(const v2f*)(xrow + k0);      // x[row, k0+koff .. +1]
        v2f b = *(const v2f*)(cp + k0);        // c broadcast (all N columns equal)
        acc = __builtin_amdgcn_wmma_f32_16x16x4_f32(
            /*neg_a=*/false, a, /*neg_b=*/false, b,
            /*c_mod=*/(short)0, acc, /*reuse_a=*/false, /*reuse_b=*/false);
    }

    // C/D layout: VGPR i -> M=i (lanes 0-15) / M=8+i (lanes 16-31), N=lane%16.
    // Column N=0 lives in lanes 0 and 16.
    if ((lane & 15) == 0) {
        const int mbase = rbase + (lane >> 4) * 8;
#pragma unroll
        for (int i = 0; i < 8; ++i)
            atomicAdd(&s[mbase + i], acc[i]);
    }
}

// ---------------------------------------------------------------------------
// Kernel C: out[b,j] = relu(x[b,j] - (s[b] - t) * c[j]), vectorized float4
// ---------------------------------------------------------------------------
__global__ void __launch_bounds__(256)
pcav_apply(const float* __restrict__ x,
           const float* __restrict__ c,
           const float* __restrict__ s,
           const float* __restrict__ tbuf,
           float* __restrict__ out) {
    const int row = blockIdx.y;
    const int j4  = blockIdx.x * blockDim.x + threadIdx.x;   // float4 index in row
    const float alpha = s[row] - tbuf[0];

    const float4 xv = ((const float4*)x)[(size_t)row * (DIMD / 4) + j4];
    const float4 cv = ((const float4*)c)[j4];
    float4 o;
    o.x = fmaxf(xv.x - alpha * cv.x, 0.0f);
    o.y = fmaxf(xv.y - alpha * cv.y, 0.0f);
    o.z = fmaxf(xv.z - alpha * cv.z, 0.0f);
    o.w = fmaxf(xv.w - alpha * cv.w, 0.0f);
    ((float4*)out)[(size_t)row * (DIMD / 4) + j4] = o;
}

// ---------------------------------------------------------------------------
extern "C" void kernel_launch(void* const* d_in, const int* in_sizes, int n_in,
                              void* d_out, int out_size, void* d_ws, size_t ws_size,
                              hipStream_t stream) {
    const float* x = (const float*)d_in[0];   // [2048, 4096]
    const float* P = (const float*)d_in[1];   // [4096, 4096] rank-1
    const float* z = (const float*)d_in[2];   // [1, 4096]
    float* out = (float*)d_out;

    float* c = (float*)d_ws;                  // 4096 floats
    float* s = c + DIMD;                      // 2048 floats
    float* t = s + BROWS;                     // 1 float

    pcav_prepare<<<1, 256, 0, stream>>>(P, z, c, s, t);
    pcav_gemv_wmma<<<dim3(BROWS / 16, KSPLIT), 32, 0, stream>>>(x, c, s);
    pcav_apply<<<dim3(DIMD / 4 / 256, BROWS), 256, 0, stream>>>(x, c, s, t, out);
}